// PostProcessor_2997887172793
// MI455X (gfx1250) — compile-verified
//
#include <hip/hip_runtime.h>
#include <math.h>

// Problem constants (fixed by the reference setup)
#define NMS_C 16
#define NMS_K 2048
#define NMS_N (NMS_C * NMS_K)
#define NT 512                      // threads per NMS block (16 wave32 waves)
#define SCORE_THRESH 0.05f
#define NMS_IOU 0.5f
#define DETS 100
#define IMG_W 1333.0f
#define IMG_H 800.0f
#define BBOX_CLIP 4.135166556742356f   // log(1000/16)

// ---------------------------------------------------------------------------
// Kernel 1: softmax prob + box decode + clip. One thread per (class,proposal).
// ---------------------------------------------------------------------------
__global__ void decode_kernel(const float* __restrict__ logit,
                              const float* __restrict__ reg,
                              const float* __restrict__ prop,
                              float* __restrict__ dec,
                              float* __restrict__ scr) {
    int i = blockIdx.x * blockDim.x + threadIdx.x;
    if (i >= NMS_N) return;

    float l0 = logit[2 * i], l1 = logit[2 * i + 1];
    float prob = 1.0f / (1.0f + expf(l0 - l1));   // softmax[:,1]
    scr[i] = prob;

    int p = i & (NMS_K - 1);                      // proposal index (tiled)
    float rx1 = prop[4 * p + 0], ry1 = prop[4 * p + 1];
    float rx2 = prop[4 * p + 2], ry2 = prop[4 * p + 3];
    float w  = rx2 - rx1 + 1.0f, h  = ry2 - ry1 + 1.0f;
    float cx = rx1 + 0.5f * w,   cy = ry1 + 0.5f * h;

    float dx = reg[8 * i + 4] / 10.0f;
    float dy = reg[8 * i + 5] / 10.0f;
    float dw = fminf(reg[8 * i + 6] / 5.0f, BBOX_CLIP);
    float dh = fminf(reg[8 * i + 7] / 5.0f, BBOX_CLIP);

    float pcx = dx * w + cx, pcy = dy * h + cy;
    float pw = expf(dw) * w, ph = expf(dh) * h;

    float x1 = pcx - 0.5f * pw;
    float y1 = pcy - 0.5f * ph;
    float x2 = pcx + 0.5f * pw - 1.0f;
    float y2 = pcy + 0.5f * ph - 1.0f;

    x1 = fminf(fmaxf(x1, 0.0f), IMG_W - 1.0f);
    y1 = fminf(fmaxf(y1, 0.0f), IMG_H - 1.0f);
    x2 = fminf(fmaxf(x2, 0.0f), IMG_W - 1.0f);
    y2 = fminf(fmaxf(y2, 0.0f), IMG_H - 1.0f);

    float4 b = {x1, y1, x2, y2};
    *(float4*)(dec + 4 * (size_t)i) = b;
}

// ---------------------------------------------------------------------------
// Kernel 2: per-class greedy NMS, one workgroup per class, fully LDS-resident.
// CDNA5 async global->LDS (ASYNCcnt) stages the score-sorted boxes; the greedy
// loop only barriers on iterations that actually write LDS (kept boxes).
// ---------------------------------------------------------------------------
__global__ void __launch_bounds__(NT)
nms_kernel(const float* __restrict__ dec, const float* __restrict__ scr,
           int* __restrict__ keep) {
    __shared__ float s_score[NMS_K];       //  8 KB   masked score (=-inf invalid)
    __shared__ int   s_idx[NMS_K];         //  8 KB   original index in class
    __shared__ float s_box[NMS_K * 4];     // 32 KB   sorted boxes
    __shared__ unsigned char s_supp[NMS_K];//  2 KB
    __shared__ unsigned char s_keep[NMS_K];//  2 KB

    const int c = blockIdx.x;
    const int tid = threadIdx.x;

    // 1. keys: score if valid else -inf (matches jnp.where(valid, s, -inf))
    for (int j = tid; j < NMS_K; j += NT) {
        float s = scr[(size_t)c * NMS_K + j];
        s_score[j] = (s > SCORE_THRESH) ? s : -INFINITY;
        s_idx[j] = j;
        s_supp[j] = 0;
        s_keep[j] = 0;
    }
    __syncthreads();

    // 2. bitonic sort, descending by key (distinct keys => order matches argsort)
    for (int k = 2; k <= NMS_K; k <<= 1) {
        for (int jj = k >> 1; jj > 0; jj >>= 1) {
            for (int i = tid; i < NMS_K; i += NT) {
                int ixj = i ^ jj;
                if (ixj > i) {
                    bool up = ((i & k) == 0);
                    float a = s_score[i], b = s_score[ixj];
                    bool sw = up ? (a < b) : (a > b);
                    if (sw) {
                        s_score[i] = b; s_score[ixj] = a;
                        int t = s_idx[i]; s_idx[i] = s_idx[ixj]; s_idx[ixj] = t;
                    }
                }
            }
            __syncthreads();
        }
    }

    // 3. gather sorted boxes straight into LDS via CDNA5 async-to-LDS path.
    //    Each lane issues a B128 (one box) async copy; completion tracked by
    //    ASYNCcnt, drained with s_wait_asynccnt before the barrier.
    for (int j = tid; j < NMS_K; j += NT) {
        const float* gp = dec + 4 * (size_t)((size_t)c * NMS_K + s_idx[j]);
        unsigned lds_off = (unsigned)(size_t)(&s_box[4 * j]);
        asm volatile("global_load_async_to_lds_b128 %0, %1, off"
                     :: "v"(lds_off), "v"(gp) : "memory");
    }
    asm volatile("s_wait_asynccnt 0" ::: "memory");
    __syncthreads();

    // 4. greedy NMS (forward suppression == reference's "any kept j<i").
    //    Iterations that keep nothing perform no LDS writes, so the barrier is
    //    only needed (and taken) on kept iterations; the kept/break decisions
    //    are computed identically by all threads from stable LDS state.
    for (int i = 0; i < NMS_K; ++i) {
        float si = s_score[i];
        if (si <= SCORE_THRESH) break;          // sorted: tail is all invalid
        if (s_supp[i]) continue;                // suppressed: no writes, no barrier
        if (tid == 0) s_keep[i] = 1;
        float x1 = s_box[4 * i + 0], y1 = s_box[4 * i + 1];
        float x2 = s_box[4 * i + 2], y2 = s_box[4 * i + 3];
        float ai = (x2 - x1) * (y2 - y1);
        for (int j = i + 1 + tid; j < NMS_K; j += NT) {
            if (s_supp[j]) continue;
            float bx1 = s_box[4 * j + 0], by1 = s_box[4 * j + 1];
            float bx2 = s_box[4 * j + 2], by2 = s_box[4 * j + 3];
            float aj = (bx2 - bx1) * (by2 - by1);
            float iw = fmaxf(fminf(x2, bx2) - fmaxf(x1, bx1), 0.0f);
            float ih = fmaxf(fminf(y2, by2) - fmaxf(y1, by1), 0.0f);
            float inter = iw * ih;
            float iou = inter / (ai + aj - inter + 1e-12f);
            if (iou > NMS_IOU) s_supp[j] = 1;
        }
        __syncthreads();
    }
    __syncthreads();   // make s_keep / s_supp visible for writeback

    // 5. scatter keep flags back to original ordering
    for (int j = tid; j < NMS_K; j += NT)
        keep[(size_t)c * NMS_K + s_idx[j]] = (int)s_keep[j];
}

// ---------------------------------------------------------------------------
// Kernel 3: exact kth (=100th) largest kept score via binary search over the
// positive-float bit pattern (monotone). Writes -inf if n_det <= 100.
// ---------------------------------------------------------------------------
__device__ __forceinline__ void block_count(int v, int* s_cnt, int tid) {
    // wave32-level tree reduction, one shared atomic per wave
    for (int off = 16; off > 0; off >>= 1) v += __shfl_down(v, off, 32);
    if ((tid & 31) == 0) atomicAdd(s_cnt, v);
}

__global__ void __launch_bounds__(1024)
topk_kernel(const float* __restrict__ scr, const int* __restrict__ keep,
            float* __restrict__ kth_out) {
    __shared__ int s_cnt;
    const int tid = threadIdx.x;

    int lc = 0;
    for (int i = tid; i < NMS_N; i += 1024) lc += keep[i];
    if (tid == 0) s_cnt = 0;
    __syncthreads();
    block_count(lc, &s_cnt, tid);
    __syncthreads();
    int n_det = s_cnt;                 // uniform across block

    if (n_det <= DETS) {
        if (tid == 0) *kth_out = -INFINITY;
        return;
    }

    unsigned lo = 0u, hi = 0x3F800000u;   // scores are in (0,1)
    while (lo < hi) {
        unsigned mid = lo + (hi - lo + 1u) / 2u;
        __syncthreads();
        if (tid == 0) s_cnt = 0;
        __syncthreads();
        int cc = 0;
        for (int i = tid; i < NMS_N; i += 1024)
            cc += (keep[i] && (__float_as_uint(scr[i]) >= mid)) ? 1 : 0;
        block_count(cc, &s_cnt, tid);
        __syncthreads();
        if (s_cnt >= DETS) lo = mid; else hi = mid - 1u;
    }
    if (tid == 0) *kth_out = __uint_as_float(lo);
}

// ---------------------------------------------------------------------------
// Kernel 4: final masking + output assembly (out5 | labels | keep_final).
// ---------------------------------------------------------------------------
__global__ void finalize_kernel(const float* __restrict__ dec,
                                const float* __restrict__ scr,
                                const int* __restrict__ keep,
                                const float* __restrict__ kth_ptr,
                                const int* __restrict__ ulabels,
                                float* __restrict__ out) {
    int i = blockIdx.x * blockDim.x + threadIdx.x;
    if (i >= NMS_N) return;
    float kth = *kth_ptr;              // -inf => no top-k cut
    float s = scr[i];
    bool kf = (keep[i] != 0) && (s >= kth);
    float4 b = *(const float4*)(dec + 4 * (size_t)i);
    float* o5 = out + 5 * (size_t)i;
    o5[0] = kf ? b.x : 0.0f;
    o5[1] = kf ? b.y : 0.0f;
    o5[2] = kf ? b.z : 0.0f;
    o5[3] = kf ? b.w : 0.0f;
    o5[4] = kf ? s   : 0.0f;
    out[(size_t)NMS_N * 5 + i] = (float)ulabels[i / NMS_K];
    out[(size_t)NMS_N * 6 + i] = kf ? 1.0f : 0.0f;
}

// ---------------------------------------------------------------------------
extern "C" void kernel_launch(void* const* d_in, const int* in_sizes, int n_in,
                              void* d_out, int out_size, void* d_ws, size_t ws_size,
                              hipStream_t stream) {
    const float* logit  = (const float*)d_in[0];   // (C*K, 2)
    const float* reg    = (const float*)d_in[1];   // (C*K, 8)
    const float* prop   = (const float*)d_in[2];   // (K, 4)
    const int*   ulab   = (const int*)d_in[3];     // (C,)
    float* out = (float*)d_out;

    // workspace carve-out (~0.79 MB)
    char* w = (char*)d_ws;
    float* dec = (float*)w;                        // N*4 floats
    float* scr = dec + (size_t)NMS_N * 4;          // N floats
    int*   kp  = (int*)(scr + NMS_N);              // N ints
    float* kth = (float*)(kp + NMS_N);             // 1 float

    decode_kernel  <<<NMS_N / 256, 256, 0, stream>>>(logit, reg, prop, dec, scr);
    nms_kernel     <<<NMS_C, NT, 0, stream>>>(dec, scr, kp);
    topk_kernel    <<<1, 1024, 0, stream>>>(scr, kp, kth);
    finalize_kernel<<<NMS_N / 256, 256, 0, stream>>>(dec, scr, kp, kth, ulab, out);
}